// cluster_MixStyle_45878840656157
// MI455X (gfx1250) — compile-verified
//
#include <hip/hip_runtime.h>
#include <hip/hip_bf16.h>

typedef __attribute__((ext_vector_type(2))) float v2f;
typedef __attribute__((ext_vector_type(4))) float v4f;
typedef __attribute__((ext_vector_type(8))) float v8f;

#define BB   64
#define CC   64
#define HW   12544
#define KCL  8
#define NSPF 12544.0f
#define EPSF 1e-6f

// ---------------------------------------------------------------------------
// Kernel 1: per-(b,c) spatial sums s1 = sum(x), s2 = sum(x*x).
// One block per (b,c); coalesced b128 streaming + LDS tree reduction.
// ---------------------------------------------------------------------------
__global__ void mixstyle_stats_kernel(const float* __restrict__ x,
                                      float* __restrict__ s1,
                                      float* __restrict__ s2) {
  __shared__ float r1[256];
  __shared__ float r2[256];
  const int bc = blockIdx.x;
  const v4f* __restrict__ xp = (const v4f*)(x + (size_t)bc * HW);
  const int n4 = HW / 4;  // 3136
  float a1 = 0.f, a2 = 0.f;
  for (int i = threadIdx.x; i < n4; i += 256) {
    v4f v = xp[i];
    a1 += v.x + v.y + v.z + v.w;
    a2 = fmaf(v.x, v.x, a2);
    a2 = fmaf(v.y, v.y, a2);
    a2 = fmaf(v.z, v.z, a2);
    a2 = fmaf(v.w, v.w, a2);
  }
  r1[threadIdx.x] = a1;
  r2[threadIdx.x] = a2;
  __syncthreads();
  for (int s = 128; s > 0; s >>= 1) {
    if (threadIdx.x < (unsigned)s) {
      r1[threadIdx.x] += r1[threadIdx.x + s];
      r2[threadIdx.x] += r2[threadIdx.x + s];
    }
    __syncthreads();
  }
  if (threadIdx.x == 0) {
    s1[bc] = r1[0];
    s2[bc] = r2[0];
  }
}

// ---------------------------------------------------------------------------
// Kernel 2 (single block, 256 threads = 8 waves):
//   - argmax cluster id per sample
//   - segment-sum over clusters as a one-hot GEMM via V_WMMA_F32_16X16X4_F32:
//       cs[q][k][c] = sum_b onehot[k][b] * s_q[b][c]    (M=16,N=16,K=64)
//   - cluster mu/std, then per-(b,c) scale/bias for the affine apply pass.
// ---------------------------------------------------------------------------
__global__ void mixstyle_cluster_kernel(const float* __restrict__ cmap,   // [B*K]
                                        const float* __restrict__ lmda,   // [B]
                                        const float* __restrict__ s1,     // [B*C]
                                        const float* __restrict__ s2,     // [B*C]
                                        float* __restrict__ scale,        // [B*C]
                                        float* __restrict__ bias) {       // [B*C]
  __shared__ int   cid[BB];
  __shared__ float lmdaL[BB];
  __shared__ float cs[2][KCL][CC];
  __shared__ float cmuL[KCL][CC];
  __shared__ float cstdL[KCL][CC];

  const int t = threadIdx.x;
  if (t < BB) {
    const float* row = cmap + t * KCL;
    float best = row[0];
    int bi = 0;
    for (int k = 1; k < KCL; ++k) {
      float v = row[k];
      if (v > best) { best = v; bi = k; }  // first max wins, as jnp.argmax
    }
    cid[t] = bi;
    lmdaL[t] = lmda[t];
  }
  __syncthreads();

  // --- WMMA segment-sum (all 256 lanes active: EXEC all ones) ---
  {
    const int wv   = t >> 5;        // wave 0..7
    const int lane = t & 31;
    const int q    = wv >> 2;       // 0 -> s1, 1 -> s2
    const int c0   = (wv & 3) * 16; // channel tile
    const float* __restrict__ src = q ? s2 : s1;
    const int half = lane >> 4;     // K half select (A: K=0,1 vs K=2,3)
    const int mn   = lane & 15;     // A row / B column within tile

    v8f acc = {};
#pragma unroll
    for (int kk0 = 0; kk0 < BB; kk0 += 4) {
      const int kb = kk0 + half * 2;
      v2f a, b;
      a.x = (cid[kb + 0] == mn) ? 1.0f : 0.0f;
      a.y = (cid[kb + 1] == mn) ? 1.0f : 0.0f;
      b.x = src[(kb + 0) * CC + c0 + mn];
      b.y = src[(kb + 1) * CC + c0 + mn];
      acc = __builtin_amdgcn_wmma_f32_16x16x4_f32(
          /*neg_a=*/false, a, /*neg_b=*/false, b,
          /*c_mod=*/(short)0, acc, /*reuse_a=*/false, /*reuse_b=*/false);
    }
    // D layout: vgpr i, lanes 0-15 -> (M=i, N=lane). Clusters are rows 0..7.
    if (lane < 16) {
#pragma unroll
      for (int i = 0; i < 8; ++i) cs[q][i][c0 + lane] = acc[i];
    }
  }
  __syncthreads();

  // --- cluster mu / std ---
  for (int idx = t; idx < KCL * CC; idx += 256) {
    const int k = idx / CC, c = idx % CC;
    int cnt = 0;
    for (int b = 0; b < BB; ++b) cnt += (cid[b] == k) ? 1 : 0;
    const float nk  = fmaxf((float)cnt * NSPF, 1.0f);
    const float mu  = cs[0][k][c] / nk;
    const float var = (cs[1][k][c] - nk * mu * mu) / fmaxf(nk - 1.0f, 1.0f);
    cmuL[k][c]  = mu;
    cstdL[k][c] = sqrtf(var + EPSF);
  }
  __syncthreads();

  // --- per-(b,c) affine coefficients ---
  for (int idx = t; idx < BB * CC; idx += 256) {
    const int b = idx / CC, c = idx % CC;
    const float m1   = s1[idx] / NSPF;
    const float var  = (s2[idx] - NSPF * m1 * m1) / (NSPF - 1.0f);
    const float sstd = sqrtf(var + EPSF);
    const float l    = lmdaL[b];
    const int   k    = cid[b];
    const float r    = cstdL[k][c] / sstd;
    scale[idx] = l + (1.0f - l) * r;
    bias[idx]  = (1.0f - l) * (cmuL[k][c] - m1 * r);
  }
}

// ---------------------------------------------------------------------------
// Kernel 3: out = x * scale[bc] + bias[bc].  Coalesced b128 traffic.
// Non-temporal b128 stores so the 205MB output stream does not evict x
// (mostly L2-resident after pass 1; L2 = 192MB) before we finish reading it.
// ---------------------------------------------------------------------------
__global__ void mixstyle_apply_kernel(const float* __restrict__ x,
                                      const float* __restrict__ scale,
                                      const float* __restrict__ bias,
                                      float* __restrict__ out) {
  const int i4 = blockIdx.x * 256 + threadIdx.x;
  const int bc = i4 / (HW / 4);
  const float sc = scale[bc];
  const float bs = bias[bc];
  const v4f* __restrict__ xp = (const v4f*)x;
  v4f v = xp[i4];
  v4f o;
  o.x = fmaf(v.x, sc, bs);
  o.y = fmaf(v.y, sc, bs);
  o.z = fmaf(v.z, sc, bs);
  o.w = fmaf(v.w, sc, bs);
  __builtin_nontemporal_store(o, (v4f*)out + i4);
}

// ---------------------------------------------------------------------------
extern "C" void kernel_launch(void* const* d_in, const int* in_sizes, int n_in,
                              void* d_out, int out_size, void* d_ws, size_t ws_size,
                              hipStream_t stream) {
  const float* x    = (const float*)d_in[0];  // [64,64,112,112]
  const float* cmap = (const float*)d_in[1];  // [1,64,8]
  const float* lmda = (const float*)d_in[2];  // [64,1,1,1]
  float* out = (float*)d_out;

  float* w     = (float*)d_ws;
  float* s1    = w;            // 4096
  float* s2    = w + 4096;     // 4096
  float* scale = w + 8192;     // 4096
  float* bias  = w + 12288;    // 4096

  // Pass 1: per-(b,c) sums
  mixstyle_stats_kernel<<<BB * CC, 256, 0, stream>>>(x, s1, s2);
  // Pass 2: cluster stats (WMMA segment-sum) + affine coefficients
  mixstyle_cluster_kernel<<<1, 256, 0, stream>>>(cmap, lmda, s1, s2, scale, bias);
  // Pass 3: streaming affine apply
  const int total4 = (BB * CC * HW) / 4;      // 12,845,056 float4
  mixstyle_apply_kernel<<<total4 / 256, 256, 0, stream>>>(x, scale, bias, out);
}